// CDSNetwork_48722109006622
// MI455X (gfx1250) — compile-verified
//
#include <hip/hip_runtime.h>
#include <hip/hip_bf16.h>

typedef float v2f __attribute__((ext_vector_type(2)));
typedef float v4f __attribute__((ext_vector_type(4)));
typedef float v8f __attribute__((ext_vector_type(8)));

// Problem constants (from reference)
#define OBS_DIM   128
#define ACTION_DIM 16
#define N_AGENTS   8
#define HIDDEN   512
#define ASD      256
#define ASH      256
#define M_ROWS  (32 * 256 * 8)      // 65536

#define BM 128
#define KC 32
#define INVALID_IDX 0xFFFFFFFFu
#define MAXPADM (M_ROWS + N_AGENTS * BM)     // 66560, multiple of 128
#define MAXBLK  (MAXPADM / BM)               // 520

// ---------------------------------------------------------------------------
// Async global->LDS copy (CDNA5 GLOBAL_LOAD_ASYNC_TO_LDS_*, ASYNCcnt-tracked)
// Builtin signature (from hipcc diagnostic): b128 takes
//   (int __vector(4) addrspace(1)*, int __vector(4) addrspace(3)*, Imm, Imm)
// ---------------------------------------------------------------------------
#if __has_builtin(__builtin_amdgcn_global_load_async_to_lds_b128) && \
    __has_builtin(__builtin_amdgcn_global_load_async_to_lds_b64)
#define ASYNC_LDS 1
#else
#define ASYNC_LDS 0
#endif

typedef int vi4 __attribute__((vector_size(16)));
typedef int vi2 __attribute__((vector_size(8)));
using gvi4 = __attribute__((address_space(1))) vi4;
using lvi4 = __attribute__((address_space(3))) vi4;
using gvi2 = __attribute__((address_space(1))) vi2;
using lvi2 = __attribute__((address_space(3))) vi2;

template <int BYTES>
__device__ __forceinline__ void cp_async(const float* g, float* l) {
#if ASYNC_LDS
  // generic-shared addr[31:0] is the LDS offset (aperture mapping), so an
  // integer cast to the 32-bit as(3) pointer is the LDS address.
  unsigned long long ga = (unsigned long long)g;
  unsigned           la = (unsigned)(unsigned long long)l;
  if constexpr (BYTES == 16)
    __builtin_amdgcn_global_load_async_to_lds_b128((gvi4*)ga, (lvi4*)la, 0, 0);
  else
    __builtin_amdgcn_global_load_async_to_lds_b64((gvi2*)ga, (lvi2*)la, 0, 0);
#else
  if constexpr (BYTES == 16)
    *(v4f*)l = *(const v4f*)g;
  else
    *(v2f*)l = *(const v2f*)g;
#endif
}

template <int N>
__device__ __forceinline__ void wait_async() {
#if ASYNC_LDS
#if __has_builtin(__builtin_amdgcn_s_wait_asynccnt)
  __builtin_amdgcn_s_wait_asynccnt(N);
#else
  asm volatile("s_wait_asynccnt %0" ::"n"(N) : "memory");
#endif
#endif
}

// ---------------------------------------------------------------------------
// Prep kernels: counting sort of rows by agent id into 128-aligned segments
// ---------------------------------------------------------------------------
__global__ __launch_bounds__(256) void prep_init(unsigned* __restrict__ idx,
                                                 int* __restrict__ hist,
                                                 int* __restrict__ tile_agent) {
  int t = blockIdx.x * blockDim.x + threadIdx.x;
  if (t < MAXPADM) idx[t] = INVALID_IDX;
  if (t < N_AGENTS) hist[t] = 0;
  if (t < MAXBLK) tile_agent[t] = 0;
}

__global__ __launch_bounds__(256) void prep_hist(const int* __restrict__ ids,
                                                 int* __restrict__ hist) {
  int t = blockIdx.x * blockDim.x + threadIdx.x;
  if (t < M_ROWS) atomicAdd(&hist[ids[t] & (N_AGENTS - 1)], 1);
}

__global__ void prep_plan(const int* __restrict__ hist,
                          int* __restrict__ cursor,
                          int* __restrict__ tile_agent) {
  if (blockIdx.x == 0 && threadIdx.x == 0) {
    int run = 0;
    for (int a = 0; a < N_AGENTS; ++a) {
      cursor[a] = run;
      int cnt = hist[a];
      int blocks = (cnt + BM - 1) / BM;
      int b0 = run / BM;
      for (int b = 0; b < blocks; ++b) tile_agent[b0 + b] = a;
      run += blocks * BM;
    }
  }
}

__global__ __launch_bounds__(256) void prep_scatter(const int* __restrict__ ids,
                                                    int* __restrict__ cursor,
                                                    unsigned* __restrict__ idx) {
  int t = blockIdx.x * blockDim.x + threadIdx.x;
  if (t < M_ROWS) {
    int a = ids[t] & (N_AGENTS - 1);
    int pos = atomicAdd(&cursor[a], 1);
    idx[pos] = (unsigned)t;
  }
}

// ---------------------------------------------------------------------------
// Fused WMMA fp32 GEMM:  C = act(Asrc @ B + bias)
//   BM=128 rows/block, BN cols/block, KC=32 K-slab, 8 waves, wave -> 16xBN.
//   Double-buffered LDS, async global->LDS staging (ASYNCcnt pipelined).
//   GATHER : A row g = idx[m] (INVALID -> row 0, result discarded)
//   SCATTER: store row = idx[m] (INVALID -> dropped)
//   AGENT  : B/bias base offset by tile_agent[blockIdx.y]
//   Split-K A source: k < K0 reads A0, else A1 (combined = [shared | af])
// ---------------------------------------------------------------------------
template <int BN, bool RELU, bool GATHER, bool SCATTER, bool AGENT>
__global__ __launch_bounds__(256) void wmma_gemm(
    const float* __restrict__ A0, int lda0, int K0,
    const float* __restrict__ A1, int lda1,
    const float* __restrict__ Bw, int N, int K,
    const float* __restrict__ bias,
    float* __restrict__ C, int ldc,
    const unsigned* __restrict__ gidx,
    const int* __restrict__ tile_agent) {
  __shared__ float As[2][BM][KC + 4];   // +4 floats: 16B-aligned rows, bank spread
  __shared__ float Bs[2][KC][BN + 4];
  __shared__ unsigned rowIdx[BM];

  const int tid  = threadIdx.x;
  const int wave = tid >> 5;
  const int lane = tid & 31;
  const int hl   = lane & 15;
  const int hi   = lane >> 4;            // lane half selects K pair / M+8
  const int m0   = blockIdx.y * BM;
  const int n0   = blockIdx.x * BN;

  int agent = 0;
  if (AGENT) agent = tile_agent[blockIdx.y];
  const float* Bbase    = Bw + (size_t)agent * K * N;
  const float* biasBase = bias + (size_t)(AGENT ? agent * N : 0);

  if (GATHER || SCATTER) {
    if (tid < BM) rowIdx[tid] = gidx[m0 + tid];
  }
  __syncthreads();   // rowIdx visible to all stagers

  // async ops issued per thread per stage (uniform across all waves!)
  constexpr int PER_STAGE = 4 + (BN == 64 ? 2 : 1);

  auto stage = [&](int kt, int b) {
    const int k0 = kt * KC;
    const bool useA1 = (A1 != nullptr) && (k0 >= K0);
    const float* Asrc = useA1 ? A1 : A0;
    const int    lda  = useA1 ? lda1 : lda0;
    const int    acol = useA1 ? (k0 - K0) : k0;
    // A slab: 128 x 32 floats = 1024 x 16B chunks, 4 per thread
#pragma unroll
    for (int i = 0; i < 4; ++i) {
      int linear = tid + i * 256;
      int r  = linear >> 3;
      int c4 = (linear & 7) * 4;
      int srow = m0 + r;
      if (GATHER) {
        unsigned g = rowIdx[r];
        srow = (g == INVALID_IDX) ? 0 : (int)g;
      }
      cp_async<16>(Asrc + (size_t)srow * lda + acol + c4, &As[b][r][c4]);
    }
    // B slab: 32 x BN floats
    if (BN == 64) {
#pragma unroll
      for (int i = 0; i < 2; ++i) {
        int linear = tid + i * 256;        // 512 x 16B chunks
        int r = linear >> 4;
        int c = (linear & 15) * 4;
        cp_async<16>(Bbase + (size_t)(k0 + r) * N + n0 + c, &Bs[b][r][c]);
      }
    } else {                               // BN==16: 256 x 8B chunks, 1 each
      int r = tid >> 3;
      int c = (tid & 7) * 2;
      cp_async<8>(Bbase + (size_t)(k0 + r) * N + n0 + c, &Bs[b][r][c]);
    }
  };

  constexpr int NFRAG = BN / 16;
  v8f acc[NFRAG] = {};

  const int KT = K / KC;
  stage(0, 0);
  for (int kt = 0; kt < KT; ++kt) {
    const int cur = kt & 1;
    if (kt + 1 < KT) {
      stage(kt + 1, cur ^ 1);
      wait_async<PER_STAGE>();   // drain slab kt's copies (in-order completion)
    } else {
      wait_async<0>();
    }
    __syncthreads();             // all waves' slab-kt data visible

    const float(*A_s)[KC + 4] = As[cur];
    const float(*B_s)[BN + 4] = Bs[cur];
    // 16x16x4 fp32 WMMA micro-kernel
#pragma unroll
    for (int kk = 0; kk < KC; kk += 4) {
      const int ka = kk + hi * 2;
      v2f a = *(const v2f*)&A_s[wave * 16 + hl][ka];
#pragma unroll
      for (int f = 0; f < NFRAG; ++f) {
        v2f b;
        b.x = B_s[ka][f * 16 + hl];
        b.y = B_s[ka + 1][f * 16 + hl];
        acc[f] = __builtin_amdgcn_wmma_f32_16x16x4_f32(
            false, a, false, b, (short)0, acc[f], false, false);
      }
    }
    __syncthreads();             // done reading buf before it is restaged
  }

  // Epilogue: bias + optional ReLU, optional scatter store
#pragma unroll
  for (int f = 0; f < NFRAG; ++f) {
    const int col = n0 + f * 16 + hl;
    const float bv = biasBase[col];
#pragma unroll
    for (int r = 0; r < 8; ++r) {
      const int gr = wave * 16 + (hi ? r + 8 : r);
      float v = acc[f][r] + bv;
      if (RELU) v = fmaxf(v, 0.0f);
      if (SCATTER) {
        unsigned g = rowIdx[gr];
        if (g != INVALID_IDX) C[(size_t)g * ldc + col] = v;
      } else {
        C[(size_t)(m0 + gr) * ldc + col] = v;
      }
    }
  }
}

// ---------------------------------------------------------------------------
// Value head: value[m] = [shared|af] . Wv + bv   (one wave per row)
// ---------------------------------------------------------------------------
__global__ __launch_bounds__(256) void value_head(
    const float* __restrict__ shared_act, const float* __restrict__ af,
    const float* __restrict__ Wv, const float* __restrict__ bv,
    float* __restrict__ out) {
  const int wave = threadIdx.x >> 5;
  const int lane = threadIdx.x & 31;
  const int row  = blockIdx.x * 8 + wave;
  if (row >= M_ROWS) return;
  float s = 0.0f;
  const float* sh = shared_act + (size_t)row * HIDDEN;
#pragma unroll 4
  for (int k = lane; k < HIDDEN; k += 32) s += sh[k] * Wv[k];
  const float* a = af + (size_t)row * ASD;
#pragma unroll 4
  for (int k = lane; k < ASD; k += 32) s += a[k] * Wv[HIDDEN + k];
#pragma unroll
  for (int off = 16; off > 0; off >>= 1) s += __shfl_xor(s, off, 32);
  if (lane == 0) out[row] = s + bv[0];
}

// ---------------------------------------------------------------------------
extern "C" void kernel_launch(void* const* d_in, const int* in_sizes, int n_in,
                              void* d_out, int out_size, void* d_ws, size_t ws_size,
                              hipStream_t stream) {
  const float* obs  = (const float*)d_in[0];
  const int*   ids  = (const int*)d_in[1];
  const float* We1  = (const float*)d_in[2];
  const float* be1  = (const float*)d_in[3];
  const float* We2  = (const float*)d_in[4];
  const float* be2  = (const float*)d_in[5];
  const float* W1   = (const float*)d_in[6];
  const float* b1   = (const float*)d_in[7];
  const float* W2   = (const float*)d_in[8];
  const float* b2   = (const float*)d_in[9];
  const float* Wv   = (const float*)d_in[10];
  const float* bv   = (const float*)d_in[11];
  const float* Wp1  = (const float*)d_in[12];
  const float* bp1  = (const float*)d_in[13];
  const float* Wp2  = (const float*)d_in[14];
  const float* bp2  = (const float*)d_in[15];
  float* out = (float*)d_out;

  // Workspace layout (256B aligned regions)
  char* ws = (char*)d_ws;
  size_t off = 0;
  auto carve = [&](size_t bytes) {
    size_t r = off;
    off = (off + bytes + 255) & ~(size_t)255;
    return r;
  };
  int*      hist   = (int*)     (ws + carve(N_AGENTS * sizeof(int)));
  int*      cursor = (int*)     (ws + carve(N_AGENTS * sizeof(int)));
  int*      tagent = (int*)     (ws + carve(MAXBLK * sizeof(int)));
  unsigned* idx    = (unsigned*)(ws + carve(MAXPADM * sizeof(unsigned)));
  float*    enc1   = (float*)   (ws + carve((size_t)M_ROWS * HIDDEN * 4));   // reused as hp1
  float*    shared = (float*)   (ws + carve((size_t)M_ROWS * HIDDEN * 4));
  float*    h1     = (float*)   (ws + carve((size_t)MAXPADM * ASH * 4));
  float*    af     = (float*)   (ws + carve((size_t)M_ROWS * ASD * 4));
  float*    hp1    = enc1;      // enc1 dead after G2

  // --- agent binning ---
  prep_init<<<(MAXPADM + 255) / 256, 256, 0, stream>>>(idx, hist, tagent);
  prep_hist<<<(M_ROWS + 255) / 256, 256, 0, stream>>>(ids, hist);
  prep_plan<<<1, 32, 0, stream>>>(hist, cursor, tagent);
  prep_scatter<<<(M_ROWS + 255) / 256, 256, 0, stream>>>(ids, cursor, idx);

  // --- G1: enc1 = relu(obs @ We1 + be1)   [65536x128 @ 128x512] ---
  wmma_gemm<64, true, false, false, false>
      <<<dim3(HIDDEN / 64, M_ROWS / BM), 256, 0, stream>>>(
          obs, OBS_DIM, OBS_DIM, nullptr, 0,
          We1, HIDDEN, OBS_DIM, be1, enc1, HIDDEN, nullptr, nullptr);

  // --- G2: shared = relu(enc1 @ We2 + be2)  [K=512] ---
  wmma_gemm<64, true, false, false, false>
      <<<dim3(HIDDEN / 64, M_ROWS / BM), 256, 0, stream>>>(
          enc1, HIDDEN, HIDDEN, nullptr, 0,
          We2, HIDDEN, HIDDEN, be2, shared, HIDDEN, nullptr, nullptr);

  // --- G3: h1[p] = relu(shared[idx[p]] @ W1[agent] + b1[agent]) ---
  wmma_gemm<64, true, true, false, true>
      <<<dim3(ASH / 64, MAXBLK), 256, 0, stream>>>(
          shared, HIDDEN, HIDDEN, nullptr, 0,
          W1, ASH, HIDDEN, b1, h1, ASH, idx, tagent);

  // --- G4: af[idx[p]] = h1[p] @ W2[agent] + b2[agent]  (scatter) ---
  wmma_gemm<64, false, false, true, true>
      <<<dim3(ASD / 64, MAXBLK), 256, 0, stream>>>(
          h1, ASH, ASH, nullptr, 0,
          W2, ASD, ASH, b2, af, ASD, idx, tagent);

  // --- G5: hp1 = relu([shared|af] @ Wp1 + bp1)  split-K, K=768 ---
  wmma_gemm<64, true, false, false, false>
      <<<dim3(HIDDEN / 64, M_ROWS / BM), 256, 0, stream>>>(
          shared, HIDDEN, HIDDEN, af, ASD,
          Wp1, HIDDEN, HIDDEN + ASD, bp1, hp1, HIDDEN, nullptr, nullptr);

  // --- G6: logits = hp1 @ Wp2 + bp2   [N=16] ---
  wmma_gemm<16, false, false, false, false>
      <<<dim3(1, M_ROWS / BM), 256, 0, stream>>>(
          hp1, HIDDEN, HIDDEN, nullptr, 0,
          Wp2, ACTION_DIM, HIDDEN, bp2, out + M_ROWS, ACTION_DIM,
          nullptr, nullptr);

  // --- value head ---
  value_head<<<M_ROWS / 8, 256, 0, stream>>>(shared, af, Wv, bv, out);

  (void)in_sizes; (void)n_in; (void)out_size; (void)ws_size;
}